// ManoLayer_5669356832979
// MI455X (gfx1250) — compile-verified
//
#include <hip/hip_runtime.h>

typedef __attribute__((ext_vector_type(2))) float v2f;
typedef __attribute__((ext_vector_type(8))) float v8f;

#define NVERT 778
#define NJNT  16
#define NB    8192
#define KD    148     // padded K: betas(10) + pose_map(135) + 3 zero pad
#define XD    2334    // NVERT*3
#define XP    2336    // padded X = 146 tiles * 16
#define SC_STRIDE 152 // C tile row stride in LDS (pad)
#define VP_STRIDE 2344// v_posed row stride in LDS (pad)

__constant__ int c_tip[5]      = {745, 317, 444, 556, 673};
__constant__ int c_reorder[21] = {0,13,14,15,16,1,2,3,17,4,5,6,18,10,11,12,19,7,8,9,20};

// ---------------------------------------------------------------------------
// Pack [shapedirs | posedirs] into a K-major (KD x XP) zero-padded matrix.
// ---------------------------------------------------------------------------
__global__ void pack_dirs_k(const float* __restrict__ sd, const float* __restrict__ pd,
                            float* __restrict__ D) {
    int t = blockIdx.x * blockDim.x + threadIdx.x;
    if (t >= KD * XP) return;
    int k = t / XP, x = t % XP;
    float v = 0.f;
    if (x < XD) {
        if (k < 10)        v = sd[x * 10 + k];          // shapedirs (NV,3,10)
        else if (k < 145)  v = pd[x * 135 + (k - 10)];  // posedirs  (NV,3,135)
    }
    D[t] = v;
}

// ---------------------------------------------------------------------------
// Fold joint regressor through shapedirs/template:
//   Jsd[(j*3+d)*10+s] = sum_v Jr[j,v]*shapedirs[v,d,s];  Jvt[j*3+d] = Jr@template
// ---------------------------------------------------------------------------
__global__ void reg_pre_k(const float* __restrict__ jr, const float* __restrict__ sd,
                          const float* __restrict__ vt, float* __restrict__ jsd,
                          float* __restrict__ jvt) {
    int t = threadIdx.x;
    if (t < 480) {
        int ji = t / 30, rem = t % 30, d = rem / 10, s = rem % 10;
        float acc = 0.f;
        for (int v = 0; v < NVERT; ++v) acc += jr[ji * NVERT + v] * sd[(v * 3 + d) * 10 + s];
        jsd[t] = acc;
    } else if (t < 528) {
        int u = t - 480, ji = u / 3, d = u % 3;
        float acc = 0.f;
        for (int v = 0; v < NVERT; ++v) acc += jr[ji * NVERT + v] * vt[v * 3 + d];
        jvt[u] = acc;
    }
}

// ---------------------------------------------------------------------------
// Per (batch, joint): PCA pose -> axis-angle -> Rodrigues rotation.
// Writes rot[b,j,9] and the GEMM coefficient row C[b] = [betas | rot-I | 0pad].
// ---------------------------------------------------------------------------
__global__ void pose_rod_k(const float* __restrict__ pc, const float* __restrict__ betas,
                           const float* __restrict__ comps, const float* __restrict__ hmean,
                           float* __restrict__ rot, float* __restrict__ C) {
    int t = blockIdx.x * blockDim.x + threadIdx.x;
    if (t >= NB * NJNT) return;
    int b = t >> 4, jt = t & 15;
    float p0, p1, p2;
    if (jt == 0) {
        p0 = pc[b * 48 + 0]; p1 = pc[b * 48 + 1]; p2 = pc[b * 48 + 2];
    } else {
        int base = (jt - 1) * 3;
        float a0 = hmean[base], a1 = hmean[base + 1], a2 = hmean[base + 2];
        for (int c = 0; c < 45; ++c) {
            float w = pc[b * 48 + 3 + c];
            a0 += w * comps[c * 45 + base + 0];
            a1 += w * comps[c * 45 + base + 1];
            a2 += w * comps[c * 45 + base + 2];
        }
        p0 = a0; p1 = a1; p2 = a2;
    }
    float th  = sqrtf(p0 * p0 + p1 * p1 + p2 * p2 + 1e-8f);
    float inv = 1.0f / th;
    float kx = p0 * inv, ky = p1 * inv, kz = p2 * inv;
    float st = sinf(th), ct = cosf(th), o = 1.f - ct;
    float R[9];
    R[0] = ct + o * kx * kx;  R[1] = o * kx * ky - st * kz; R[2] = o * kx * kz + st * ky;
    R[3] = o * ky * kx + st * kz; R[4] = ct + o * ky * ky;  R[5] = o * ky * kz - st * kx;
    R[6] = o * kz * kx - st * ky; R[7] = o * kz * ky + st * kx; R[8] = ct + o * kz * kz;
    float* rp = rot + (size_t)(b * NJNT + jt) * 9;
    for (int e = 0; e < 9; ++e) rp[e] = R[e];
    if (jt == 0) {
        float* cb = C + (size_t)b * KD;
        for (int s = 0; s < 10; ++s) cb[s] = betas[b * 10 + s];
        cb[145] = 0.f; cb[146] = 0.f; cb[147] = 0.f;
    } else {
        float* cb = C + (size_t)b * KD + 10 + (jt - 1) * 9;
        for (int e = 0; e < 9; ++e) cb[e] = R[e] - ((e == 0 || e == 4 || e == 8) ? 1.f : 0.f);
    }
}

// ---------------------------------------------------------------------------
// Per batch: joints from folded regressor, kinematic chain (5 fingers of 3
// rooted at joint 0), emit T_skin (3x4 per joint) and global joint positions.
// ---------------------------------------------------------------------------
__global__ void chain_k(const float* __restrict__ rot, const float* __restrict__ betas,
                        const float* __restrict__ jsd, const float* __restrict__ jvt,
                        float* __restrict__ tskin, float* __restrict__ jtr16) {
    int b = blockIdx.x * blockDim.x + threadIdx.x;
    if (b >= NB) return;
    float bt[10];
    for (int s = 0; s < 10; ++s) bt[s] = betas[b * 10 + s];
    float j[NJNT * 3];
    for (int u = 0; u < NJNT * 3; ++u) {
        float acc = jvt[u];
        for (int s = 0; s < 10; ++s) acc += jsd[u * 10 + s] * bt[s];
        j[u] = acc;
    }
    auto wskin = [&](int ji, const float* T) {
        float* ts = tskin + (size_t)(b * NJNT + ji) * 12;
        float jx = j[ji * 3], jy = j[ji * 3 + 1], jz = j[ji * 3 + 2];
        for (int r = 0; r < 3; ++r) {
            ts[r * 4 + 0] = T[r * 4 + 0]; ts[r * 4 + 1] = T[r * 4 + 1]; ts[r * 4 + 2] = T[r * 4 + 2];
            ts[r * 4 + 3] = T[r * 4 + 3] - (T[r * 4] * jx + T[r * 4 + 1] * jy + T[r * 4 + 2] * jz);
        }
        float* jo = jtr16 + (size_t)(b * NJNT + ji) * 3;
        jo[0] = T[3]; jo[1] = T[7]; jo[2] = T[11];
    };
    float T0[12];
    const float* r0 = rot + (size_t)b * NJNT * 9;
    T0[0] = r0[0]; T0[1] = r0[1]; T0[2]  = r0[2]; T0[3]  = j[0];
    T0[4] = r0[3]; T0[5] = r0[4]; T0[6]  = r0[5]; T0[7]  = j[1];
    T0[8] = r0[6]; T0[9] = r0[7]; T0[10] = r0[8]; T0[11] = j[2];
    wskin(0, T0);
    for (int f = 0; f < 5; ++f) {
        float cur[12];
        for (int e = 0; e < 12; ++e) cur[e] = T0[e];
        int par = 0;
        for (int c = 0; c < 3; ++c) {
            int i = f * 3 + 1 + c;
            const float* ri = rot + (size_t)(b * NJNT + i) * 9;
            float tx = j[i * 3 + 0] - j[par * 3 + 0];
            float ty = j[i * 3 + 1] - j[par * 3 + 1];
            float tz = j[i * 3 + 2] - j[par * 3 + 2];
            float nxt[12];
            for (int r = 0; r < 3; ++r) {
                float a0 = cur[r * 4], a1 = cur[r * 4 + 1], a2 = cur[r * 4 + 2];
                nxt[r * 4 + 0] = a0 * ri[0] + a1 * ri[3] + a2 * ri[6];
                nxt[r * 4 + 1] = a0 * ri[1] + a1 * ri[4] + a2 * ri[7];
                nxt[r * 4 + 2] = a0 * ri[2] + a1 * ri[5] + a2 * ri[8];
                nxt[r * 4 + 3] = a0 * tx + a1 * ty + a2 * tz + cur[r * 4 + 3];
            }
            for (int e = 0; e < 12; ++e) cur[e] = nxt[e];
            wskin(i, cur);
            par = i;
        }
    }
}

// ---------------------------------------------------------------------------
// Fused pose-blend GEMM + skinning. One 16-wave block owns 16 batches:
//  phase 1: v_posed(16 x 2334) = C_tile(16x148) @ D(148x2336) + template,
//           WMMA f32 16x16x4, accumulators dumped into LDS (no HBM round-trip)
//  phase 2: per wave (one batch): W(778x16) @ T_skin(16x[12 pad 16]) via WMMA,
//           intra-wave LDS transpose, per-vertex 3x4 apply vs LDS v_posed.
// ---------------------------------------------------------------------------
__global__ void __launch_bounds__(512)
fused_gemm_skin_k(const float* __restrict__ C, const float* __restrict__ D,
                  const float* __restrict__ vt, const float* __restrict__ W,
                  const float* __restrict__ tskin, const float* __restrict__ trans,
                  float* __restrict__ verts) {
    extern __shared__ float smem[];
    float* sC  = smem;                       // 16 x SC_STRIDE
    float* sVP = sC + 16 * SC_STRIDE;        // 16 x VP_STRIDE
    float* sT  = sVP + 16 * VP_STRIDE;       // 16 waves x 16 x 17
    int lane = threadIdx.x, w = threadIdx.y;
    int tid  = w * 32 + lane;
    int b0   = blockIdx.x * 16;
    int half = lane >> 4, lm = lane & 15;

    // Stage the 16 coefficient rows (coalesced) into LDS.
    for (int i = tid; i < 16 * KD; i += 512) {
        int r = i / KD, k = i % KD;
        sC[r * SC_STRIDE + k] = C[(size_t)(b0 + r) * KD + k];
    }
    __syncthreads();

    // Preload all 37 A-fragments for this block's batch tile (reused 146x).
    v2f afr[37];
    for (int kk = 0; kk < 37; ++kk) {
        afr[kk].x = sC[lm * SC_STRIDE + kk * 4 + 2 * half];
        afr[kk].y = sC[lm * SC_STRIDE + kk * 4 + 2 * half + 1];
    }

    // Phase 1: v_posed tiles -> LDS.
    for (int xt = w; xt < 146; xt += 16) {
        int x0 = xt * 16;
        const float* dcol = D + (size_t)(2 * half) * XP + x0 + lm;
        v8f acc = {};
        for (int kk = 0; kk < 37; ++kk) {
            v2f bm;
            bm.x = dcol[0];
            bm.y = dcol[XP];
            acc = __builtin_amdgcn_wmma_f32_16x16x4_f32(false, afr[kk], false, bm,
                                                        (short)0, acc, false, false);
            dcol += 4 * (size_t)XP;
        }
        int x = x0 + lm;
        float tv = (x < XD) ? vt[x] : 0.f;
        for (int r = 0; r < 8; ++r)
            sVP[(half * 8 + r) * VP_STRIDE + x] = acc[r] + tv;
    }
    __syncthreads();

    // Phase 2: wave w skins batch b0 + w.
    int b = b0 + w;
    v2f bf[4];
    for (int q = 0; q < 4; ++q) {
        int jrow = q * 4 + 2 * half;
        bf[q].x = (lm < 12) ? tskin[(size_t)(b * NJNT + jrow + 0) * 12 + lm] : 0.f;
        bf[q].y = (lm < 12) ? tskin[(size_t)(b * NJNT + jrow + 1) * 12 + lm] : 0.f;
    }
    float t0 = trans[b * 3 + 0], t1 = trans[b * 3 + 1], t2 = trans[b * 3 + 2];
    float* sTw  = sT + w * 16 * 17;
    float* sVPw = sVP + w * VP_STRIDE;
    for (int vtile = 0; vtile < 49; ++vtile) {
        int v0 = vtile * 16;
        v8f acc = {};
        for (int q = 0; q < 4; ++q) {
            int vrow = v0 + lm;
            int jc   = q * 4 + 2 * half;
            v2f a;
            a.x = (vrow < NVERT) ? W[vrow * NJNT + jc + 0] : 0.f;
            a.y = (vrow < NVERT) ? W[vrow * NJNT + jc + 1] : 0.f;
            acc = __builtin_amdgcn_wmma_f32_16x16x4_f32(false, a, false, bf[q],
                                                        (short)0, acc, false, false);
        }
        for (int r = 0; r < 8; ++r) sTw[(half * 8 + r) * 17 + lm] = acc[r];
        asm volatile("s_wait_dscnt 0" ::: "memory");  // intra-wave LDS RAW
        if (lane < 16) {
            int v = v0 + lane;
            if (v < NVERT) {
                float vp0 = sVPw[v * 3 + 0];
                float vp1 = sVPw[v * 3 + 1];
                float vp2 = sVPw[v * 3 + 2];
                const float* m = sTw + lane * 17;
                float* o = verts + (size_t)(b * NVERT + v) * 3;
                o[0] = (m[0] * vp0 + m[1] * vp1 + m[2]  * vp2 + m[3]  + t0) * 1000.f;
                o[1] = (m[4] * vp0 + m[5] * vp1 + m[6]  * vp2 + m[7]  + t1) * 1000.f;
                o[2] = (m[8] * vp0 + m[9] * vp1 + m[10] * vp2 + m[11] + t2) * 1000.f;
            }
        }
    }
}

// ---------------------------------------------------------------------------
// Final joints: reorder 16 chain joints + 5 fingertip vertices (already
// trans-shifted and scaled in verts output).
// ---------------------------------------------------------------------------
__global__ void jtr_k(const float* __restrict__ jtr16, const float* __restrict__ trans,
                      const float* __restrict__ verts, float* __restrict__ oj) {
    int t = blockIdx.x * blockDim.x + threadIdx.x;
    if (t >= NB * 21) return;
    int b = t / 21, i = t % 21;
    int r = c_reorder[i];
    float x, y, z;
    if (r < 16) {
        x = (jtr16[(size_t)(b * NJNT + r) * 3 + 0] + trans[b * 3 + 0]) * 1000.f;
        y = (jtr16[(size_t)(b * NJNT + r) * 3 + 1] + trans[b * 3 + 1]) * 1000.f;
        z = (jtr16[(size_t)(b * NJNT + r) * 3 + 2] + trans[b * 3 + 2]) * 1000.f;
    } else {
        const float* vv = verts + (size_t)(b * NVERT + c_tip[r - 16]) * 3;
        x = vv[0]; y = vv[1]; z = vv[2];
    }
    float* o = oj + (size_t)t * 3;
    o[0] = x; o[1] = y; o[2] = z;
}

extern "C" void kernel_launch(void* const* d_in, const int* in_sizes, int n_in,
                              void* d_out, int out_size, void* d_ws, size_t ws_size,
                              hipStream_t stream) {
    (void)in_sizes; (void)n_in; (void)out_size; (void)ws_size;
    const float* pc      = (const float*)d_in[0];  // (8192,48)
    const float* betas   = (const float*)d_in[1];  // (8192,10)
    const float* trans   = (const float*)d_in[2];  // (8192,3)
    const float* comps   = (const float*)d_in[3];  // (45,45)
    const float* hmean   = (const float*)d_in[4];  // (1,45)
    const float* sdirs   = (const float*)d_in[5];  // (778,3,10)
    const float* pdirs   = (const float*)d_in[6];  // (778,3,135)
    const float* vtempl  = (const float*)d_in[7];  // (1,778,3)
    const float* jregr   = (const float*)d_in[8];  // (16,778)
    const float* weights = (const float*)d_in[9];  // (778,16)

    float* verts_out = (float*)d_out;                        // (8192,778,3)
    float* jtr_out   = verts_out + (size_t)NB * NVERT * 3;   // (8192,21,3)

    float* ws = (float*)d_ws;
    size_t off = 0;
    float* Dpad  = ws + off; off += (size_t)KD * XP;
    float* Jsd   = ws + off; off += 480;
    float* Jvt   = ws + off; off += 48;
    float* C     = ws + off; off += (size_t)NB * KD;
    float* rot   = ws + off; off += (size_t)NB * NJNT * 9;
    float* tskin = ws + off; off += (size_t)NB * NJNT * 12;
    float* jtr16 = ws + off; off += (size_t)NB * NJNT * 3;

    pack_dirs_k<<<(KD * XP + 255) / 256, 256, 0, stream>>>(sdirs, pdirs, Dpad);
    reg_pre_k<<<1, 544, 0, stream>>>(jregr, sdirs, vtempl, Jsd, Jvt);
    pose_rod_k<<<(NB * NJNT + 127) / 128, 128, 0, stream>>>(pc, betas, comps, hmean, rot, C);
    chain_k<<<NB / 64, 64, 0, stream>>>(rot, betas, Jsd, Jvt, tskin, jtr16);

    size_t smem_bytes = (size_t)(16 * SC_STRIDE + 16 * VP_STRIDE + 16 * 16 * 17) * sizeof(float);
    fused_gemm_skin_k<<<NB / 16, dim3(32, 16), smem_bytes, stream>>>(
        C, Dpad, vtempl, weights, tskin, trans, verts_out);

    jtr_k<<<(NB * 21 + 255) / 256, 256, 0, stream>>>(jtr16, trans, verts_out, jtr_out);
}